// DynamicScatter_69922067579635
// MI455X (gfx1250) — compile-verified
//
#include <hip/hip_runtime.h>
#include <hip/hip_bf16.h>

// ---------------------------------------------------------------------------
// DynamicScatter (hard voxelization) for MI455X / gfx1250.
// Memory-bound scatter: no matrix math -> no WMMA. Uses CDNA5-specific paths:
//   * global_load_async_to_lds_b32 + s_wait_asynccnt  (ASYNCcnt pipeline) to
//     stage the stride-5 AoS point array into LDS with coalesced loads
//   * global_prefetch_b8 via __builtin_prefetch on the streamed key array
// ---------------------------------------------------------------------------

#define GX 352
#define GY 400
#define GZ 8
#define NBATCH 4
#define MAXVOX 40000
#define MAXPTS 48
#define PB (GX * GY * GZ)          // cells per batch = 1,126,400
#define NCELLS (NBATCH * PB)       // 4,505,600
#define SCAN_TILE 4096             // divides PB (PB/4096 = 275) -> no straddle
#define NTILES (NCELLS / SCAN_TILE)      // 1100
#define TILES_PER_BATCH (PB / SCAN_TILE) // 275
#define DUMP (NBATCH * MAXVOX)     // 160,000 voxel rows

// ---------------- CDNA5 async global->LDS helpers (inline asm) -------------
__device__ __forceinline__ void async_load_b32_to_lds(unsigned lds_byte_addr,
                                                      unsigned voff_bytes,
                                                      const void* sbase) {
  // GVS mode: copies 4 bytes from the 64-bit SGPR-pair base plus the 32-bit offset in vaddr to the LDS address in vdst
  asm volatile("global_load_async_to_lds_b32 %0, %1, %2"
               :
               : "v"(lds_byte_addr), "v"(voff_bytes), "s"(sbase)
               : "memory");
}
__device__ __forceinline__ void wait_async0() {
  asm volatile("s_wait_asynccnt 0" ::: "memory");
}

__device__ __forceinline__ int clampi(int v, int lo, int hi) {
  return v < lo ? lo : (v > hi ? hi : v);
}

// ---------------- zero-fill (harness poisons d_out/d_ws with 0xAA) ---------
__global__ __launch_bounds__(256) void k_zero4(uint4* __restrict__ p, long n16) {
  long i = (long)blockIdx.x * blockDim.x + threadIdx.x;
  long stride = (long)gridDim.x * blockDim.x;
  uint4 z = {0u, 0u, 0u, 0u};
  for (; i < n16; i += stride) p[i] = z;
}

// ---------------- pass 1: per-point key + per-cell counts ------------------
__global__ __launch_bounds__(256) void k_count(
    const float* __restrict__ pts, int N,
    unsigned long long* __restrict__ keyfeat,
    int* __restrict__ cell_count) {
  __shared__ float tile[256 * 5];
  const int tid  = threadIdx.x;
  const int base = blockIdx.x * 256;
  const float* gsrc = pts + (long)base * 5;     // tile base in global
  const int nflt = (min(256, N - base)) * 5;    // floats valid in this tile

  // low 32 bits of a generic LDS pointer == workgroup-relative LDS byte addr
  unsigned lds0 = (unsigned)(unsigned long long)(&tile[0]);
#pragma unroll
  for (int j = 0; j < 5; ++j) {
    int i = tid + j * 256;                      // coalesced across lanes
    if (i < nflt)
      async_load_b32_to_lds(lds0 + (unsigned)(i * 4), (unsigned)(i * 4), gsrc);
  }
  wait_async0();
  __syncthreads();

  const int p = base + tid;
  if (p < N) {
    float b = tile[tid * 5 + 0];
    float x = tile[tid * 5 + 1];
    float y = tile[tid * 5 + 2];
    float z = tile[tid * 5 + 3];
    float f = tile[tid * 5 + 4];
    // match reference: floor((p - pc_min) / voxel_size), then clip
    int vx = clampi((int)floorf((x - 0.0f)  / 0.2f), 0, GX - 1);
    int vy = clampi((int)floorf((y + 40.0f) / 0.2f), 0, GY - 1);
    int vz = clampi((int)floorf((z + 3.0f)  / 0.5f), 0, GZ - 1);
    int bi = (int)b;
    int key = ((bi * GX + vx) * GY + vy) * GZ + vz;
    atomicAdd(&cell_count[key], 1);
    keyfeat[p] = ((unsigned long long)__float_as_uint(f) << 32) |
                 (unsigned)key;
  }
}

// ---------------- pass 2a: per-tile occupancy scan -------------------------
__global__ __launch_bounds__(1024) void k_scan_tiles(
    const int* __restrict__ cell_count,
    int* __restrict__ cell_rank,
    int* __restrict__ tile_sums) {
  __shared__ int ssum[1024];
  const int tid = threadIdx.x;
  const long basec = (long)blockIdx.x * SCAN_TILE + (long)tid * 4;
  int4 c = *(const int4*)(cell_count + basec);
  int f0 = c.x > 0, f1 = c.y > 0, f2 = c.z > 0, f3 = c.w > 0;
  int s = f0 + f1 + f2 + f3;
  ssum[tid] = s;
  __syncthreads();
  // Hillis-Steele inclusive scan over 1024 thread sums
#pragma unroll
  for (int ofs = 1; ofs < 1024; ofs <<= 1) {
    int add = (tid >= ofs) ? ssum[tid - ofs] : 0;
    __syncthreads();
    ssum[tid] += add;
    __syncthreads();
  }
  int incl = ssum[tid];
  int excl = incl - s;
  int4 r;
  r.x = excl;
  r.y = excl + f0;
  r.z = excl + f0 + f1;
  r.w = excl + f0 + f1 + f2;
  *(int4*)(cell_rank + basec) = r;
  if (tid == 1023) tile_sums[blockIdx.x] = incl;
}

// ---------------- pass 2b: scan of tile sums, restart each batch -----------
__global__ void k_scan_blocksums(const int* __restrict__ tile_sums,
                                 int* __restrict__ tile_offs) {
  if (threadIdx.x == 0 && blockIdx.x == 0) {
    int run = 0;
    for (int t = 0; t < NTILES; ++t) {
      if ((t % TILES_PER_BATCH) == 0) run = 0;   // batch boundary restart
      tile_offs[t] = run;
      run += tile_sums[t];
    }
  }
}

// ---------------- pass 3: finalize ranks, emit coors + num_points ----------
__global__ __launch_bounds__(256) void k_finalize(
    const int* __restrict__ cell_count,
    int* __restrict__ cell_rank,
    const int* __restrict__ tile_offs,
    int* __restrict__ coors_out,
    int* __restrict__ nump_out) {
  const long c = (long)blockIdx.x * 256 + threadIdx.x;   // NCELLS exact
  int cnt = cell_count[c];
  if (cnt > 0) {
    int rank = cell_rank[c] + tile_offs[c >> 12];        // c / 4096
    cell_rank[c] = rank;                                 // final rank for pass 4
    int b = (int)(c / PB);
    if (rank < MAXVOX) {
      int row = b * MAXVOX + rank;
      nump_out[row] = cnt;
      int rem = (int)(c - (long)b * PB);
      int vz = rem & (GZ - 1);
      int vy = (rem >> 3) % GY;
      int vx = rem / (GY * GZ);
      int4 co;
      co.x = b; co.y = vz; co.z = vy; co.w = vx;         // [b, cz, cy, cx]
      *(int4*)(coors_out + (long)row * 4) = co;
    }
  }
}

// ---------------- pass 4: scatter point features ---------------------------
__global__ __launch_bounds__(256) void k_scatter(
    const unsigned long long* __restrict__ keyfeat,
    const int* __restrict__ cell_rank,
    int* __restrict__ slot_cnt,
    float* __restrict__ feat_out, int N) {
  const int p = blockIdx.x * 256 + threadIdx.x;
  if (p >= N) return;
  __builtin_prefetch(keyfeat + p + 2048, 0, 1);          // global_prefetch_b8
  unsigned long long kf = keyfeat[p];
  int key = (int)(unsigned)(kf & 0xffffffffull);
  unsigned fbits = (unsigned)(kf >> 32);
  int rank = cell_rank[key];
  int b = key / PB;
  if (rank < MAXVOX) {
    int row = b * MAXVOX + rank;
    int slot = atomicAdd(&slot_cnt[row], 1);
    if (slot < MAXPTS) {
      int rem = key - b * PB;
      int vz = rem & (GZ - 1);
      int vy = (rem >> 3) % GY;
      int vx = rem / (GY * GZ);
      float4 o;
      o.x = (float)vx; o.y = (float)vy; o.z = (float)vz;
      o.w = __uint_as_float(fbits);
      *(float4*)(feat_out + ((long)row * MAXPTS + slot) * 4) = o;
    }
  }
}

// ---------------------------------------------------------------------------
extern "C" void kernel_launch(void* const* d_in, const int* in_sizes, int n_in,
                              void* d_out, int out_size, void* d_ws, size_t ws_size,
                              hipStream_t stream) {
  const float* points = (const float*)d_in[0];
  const int N = in_sizes[0] / 5;

  // Output layout: features [DUMP,48,4] f32 | coors [DUMP,4] i32 | nump [DUMP] i32
  float* feat_out = (float*)d_out;
  int*   coors_out = (int*)((char*)d_out + (size_t)DUMP * MAXPTS * 4 * sizeof(float));
  int*   nump_out  = (int*)((char*)coors_out + (size_t)DUMP * 4 * sizeof(int));

  // Workspace layout (~45 MB), all 16B-aligned chunks
  char* w = (char*)d_ws;
  unsigned long long* keyfeat = (unsigned long long*)w; w += (size_t)N * 8;
  int* cell_count = (int*)w; w += (size_t)NCELLS * 4;
  int* cell_rank  = (int*)w; w += (size_t)NCELLS * 4;
  int* slot_cnt   = (int*)w; w += (size_t)DUMP * 4;
  int* tile_sums  = (int*)w; w += (size_t)NTILES * 4;
  int* tile_offs  = (int*)w; w += (size_t)NTILES * 4;

  // 1) zero output + counters (byte counts all divisible by 16)
  k_zero4<<<2048, 256, 0, stream>>>((uint4*)d_out,
                                    ((long)out_size * 4) / 16);
  k_zero4<<<1024, 256, 0, stream>>>((uint4*)cell_count,
                                    ((long)NCELLS * 4) / 16);
  k_zero4<<<256, 256, 0, stream>>>((uint4*)slot_cnt,
                                   ((long)DUMP * 4) / 16);

  // 2) per-point keys + dense cell counts (async-LDS staged)
  k_count<<<(N + 255) / 256, 256, 0, stream>>>(points, N, keyfeat, cell_count);

  // 3) occupancy scan -> per-batch voxel ranks
  k_scan_tiles<<<NTILES, 1024, 0, stream>>>(cell_count, cell_rank, tile_sums);
  k_scan_blocksums<<<1, 64, 0, stream>>>(tile_sums, tile_offs);

  // 4) finalize ranks; per-voxel coords + counts
  k_finalize<<<NCELLS / 256, 256, 0, stream>>>(cell_count, cell_rank, tile_offs,
                                               coors_out, nump_out);

  // 5) scatter point features into voxel slots
  k_scatter<<<(N + 255) / 256, 256, 0, stream>>>(keyfeat, cell_rank, slot_cnt,
                                                 feat_out, N);
}